// SpeedLimitReward_44787918963379
// MI455X (gfx1250) — compile-verified
//
#include <hip/hip_runtime.h>

#define HIST 4
#define CHUNK 256
static constexpr float kDT  = 0.5f;
static constexpr float kTHR = 2.23f;
static constexpr float kPI       = 3.14159265358979323846f;
static constexpr float kTWO_PI   = 6.28318530717958647692f;
static constexpr float kINV_2PI  = 0.15915494309189533577f;

typedef float v2f __attribute__((ext_vector_type(2)));
typedef float v8f __attribute__((ext_vector_type(8)));

// Low 32 bits of a generic pointer into __shared__ == LDS byte offset
// (flat LDS aperture: addr[63:32] selects LDS, addr[31:0] is the offset).
__device__ __forceinline__ unsigned lds_offset(const void* p) {
  return (unsigned)(unsigned long long)p;
}

__device__ __forceinline__ float fast_sqrtf(float x) {
#if __has_builtin(__builtin_amdgcn_sqrtf)
  return __builtin_amdgcn_sqrtf(x);   // raw v_sqrt_f32 (~1 ulp), plenty for a score
#else
  return __builtin_sqrtf(x);
#endif
}

__global__ __launch_bounds__(64)
void speed_limit_reward_kernel(const int* __restrict__ agent_ptr,
                               const float* __restrict__ aip,   // agent_infer_position (Na,Ttot,2)
                               const float* __restrict__ aih,   // agent_infer_heading  (Na,Ttot)
                               const int* __restrict__ agent_batch,
                               const float* __restrict__ pl_pos,   // (Npl,2)
                               const float* __restrict__ pl_head,  // (Npl,)
                               const int* __restrict__ pl_batch,   // sorted (Npl,)
                               const int* __restrict__ pg_type,    // (Npg,)
                               const float* __restrict__ pg_speed, // (Npg,)
                               const int* __restrict__ pg_valid,   // (Npg,) 0/1
                               const int* __restrict__ edge1,      // edge_index[1] (Npl,)
                               float* __restrict__ out,            // (B,)
                               int Npl, int Ttot, int T)
{
  __shared__ float sPos[2 * CHUNK];   // staged px,py pairs
  __shared__ float sHd[CHUNK];        // staged headings
  __shared__ int   sFl[CHUNK];        // bit0: type==0, bit1: speed-limit-valid
  __shared__ float sQx[32], sQy[32], sQyaw[32], sQspd[32];
  __shared__ int   sLo, sHi;
  __shared__ float sWavePen[2];

  const int b    = blockIdx.x;
  const int tid  = threadIdx.x;
  const int lane = tid & 31;
  const int wv   = tid >> 5;
  const int ego  = agent_ptr[b];

  if (tid == 0) {
    sWavePen[0] = 0.f; sWavePen[1] = 0.f;
    const int eb = agent_batch[ego];
    // polyline_batch is sorted -> contiguous range [lo,hi) for this batch
    int lo = 0, hi = Npl;
    while (lo < hi) { int mid = (lo + hi) >> 1; if (pl_batch[mid] <  eb) lo = mid + 1; else hi = mid; }
    sLo = lo;
    int l2 = lo, h2 = Npl;
    while (l2 < h2) { int mid = (l2 + h2) >> 1; if (pl_batch[mid] <= eb) l2 = mid + 1; else h2 = mid; }
    sHi = l2;
  }
  if (tid < T) { // stage the 32 ego queries (pos, yaw, speed)
    const size_t base = ((size_t)ego * Ttot + HIST + tid) * 2;
    const float ex = aip[base], ey = aip[base + 1];
    const float px = aip[base - 2], py = aip[base - 1];    // previous timestep
    const float dx = ex - px, dy = ey - py;
    sQx[tid]   = ex; sQy[tid] = ey;
    sQspd[tid] = __builtin_sqrtf(dx * dx + dy * dy) * (1.0f / kDT);
    sQyaw[tid] = aih[(size_t)ego * Ttot + HIST + tid];
  }
  __syncthreads();

  const int  lo = sLo, hi = sHi;
  const int  qt = wv;                       // query tile (16 timesteps) per wave
  const bool waveActive = (qt * 16 < T);

  // ---- WMMA A operand: row m = query, K = [|q|^2, qx, qy, 1] ----
  // 32-bit A 16x4 layout: lane l holds M=l%16; VGPR0 = K0 (lanes<16) / K2 (lanes>=16);
  //                       VGPR1 = K1 / K3.
  const int  mcol = lane & 15;
  const int  q0   = qt * 16 + mcol;
  const bool qin  = waveActive && (q0 < T);
  const float ex = qin ? sQx[q0] : 0.f;
  const float ey = qin ? sQy[q0] : 0.f;
  v2f Av;
  Av.x = (lane < 16) ? (ex * ex + ey * ey) : ey;
  Av.y = (lane < 16) ? ex : 1.0f;

  float yawr[8]; bool qokr[8];
  const int half8 = (lane >= 16) ? 8 : 0;
#pragma unroll
  for (int r = 0; r < 8; ++r) {
    const int qr = qt * 16 + r + half8;     // C/D row for VGPR r in this lane half
    qokr[r] = waveActive && (qr < T);
    yawr[r] = qokr[r] ? sQyaw[qr] : 0.f;
  }

  float best[8]; int bidx[8];
#pragma unroll
  for (int r = 0; r < 8; ++r) { best[r] = __builtin_inff(); bidx[r] = 0; }

  const unsigned ldsPosBase = lds_offset(&sPos[0]);
  const unsigned ldsHdBase  = lds_offset(&sHd[0]);

  for (int cb = lo; cb < hi; cb += CHUNK) {
    int nIn = hi - cb; if (nIn > CHUNK) nIn = CHUNK;

    // ---- async-stage this polyline chunk into LDS (CDNA5 async data path) ----
    for (int j = tid; j < nIn; j += 64) {
      const int i = cb + j;
      const unsigned dstPos = ldsPosBase + (unsigned)(j * 8);
      const float*   srcPos = pl_pos + (size_t)2 * i;
      asm volatile("global_load_async_to_lds_b64 %0, %1, off"
                   :: "v"(dstPos), "v"(srcPos) : "memory");
      const unsigned dstHd = ldsHdBase + (unsigned)(j * 4);
      const float*   srcHd = pl_head + i;
      asm volatile("global_load_async_to_lds_b32 %0, %1, off"
                   :: "v"(dstHd), "v"(srcHd) : "memory");
      const int e1 = edge1[i];
      sFl[j] = ((pg_type[e1] == 0) ? 1 : 0) | (pg_valid[e1] ? 2 : 0);
    }
    asm volatile("s_wait_asynccnt 0x0" ::: "memory");
    __syncthreads();

    if (waveActive) {
      const int ntiles = (nIn + 15) >> 4;
      for (int t = 0; t < ntiles; ++t) {
        const int  slot = t * 16 + (lane & 15);
        const bool inb  = slot < nIn;
        const int  ss   = inb ? slot : 0;
        const float px = sPos[2 * ss];
        const float py = sPos[2 * ss + 1];
        const float hd = sHd[ss];
        const int   fl = sFl[ss];
        const int   gidx = cb + slot;

        // B operand: col n = polyline, K = [1, -2px, -2py, |p|^2]
        v2f Bv;
        Bv.x = (lane < 16) ? 1.0f : (-2.0f * py);
        Bv.y = (lane < 16) ? (-2.0f * px) : (px * px + py * py);

#if __has_builtin(__builtin_amdgcn_wmma_f32_16x16x4_f32)
        v8f Cz = {0.f, 0.f, 0.f, 0.f, 0.f, 0.f, 0.f, 0.f};
        v8f D = __builtin_amdgcn_wmma_f32_16x16x4_f32(
            false, Av, false, Bv, (short)0, Cz, false, false);
#else
        v8f D;
#pragma unroll
        for (int r = 0; r < 8; ++r) D[r] = 0.f;  // (builtin always present per probe)
#endif
        // Branchless mask: invalid lane -> +inf addend (inf + finite == inf),
        // identical semantics to jnp.where(valid, score, inf).
        const bool  validn = inb && (fl & 1);
        const float addInf = validn ? 0.0f : __builtin_inff();
#pragma unroll
        for (int r = 0; r < 8; ++r) {
          const float d2   = fmaxf(D[r], 0.0f);   // guard expansion cancellation
          const float dist = fast_sqrtf(d2);
          const float dyaw = yawr[r] - hd;
          float tf = (dyaw + kPI) * kINV_2PI;     // floor-mod wrap to [-pi,pi)
          tf = tf - __builtin_floorf(tf);
          const float ang = __builtin_fabsf(tf * kTWO_PI - kPI);
          const float sc  = dist + 5.0f * ang + addInf;
          if (sc < best[r]) { best[r] = sc; bidx[r] = gidx; }  // first-index wins ties
        }
      }
    }
    __syncthreads();
  }

  if (waveActive) {
    float penAcc = 0.f;
#pragma unroll
    for (int r = 0; r < 8; ++r) {
      float s = best[r]; int ix = bidx[r];
#pragma unroll
      for (int off = 8; off >= 1; off >>= 1) {       // butterfly within each 16-lane half
        const float os = __shfl_xor(s, off, 32);
        const int   oi = __shfl_xor(ix, off, 32);
        if (os < s || (os == s && oi < ix)) { s = os; ix = oi; }
      }
      if (((lane & 15) == 0) && qokr[r]) {           // lane 0 -> row r, lane 16 -> row r+8
        const int   qq  = qt * 16 + r + half8;
        const int   e1  = edge1[ix];
        const float csl = pg_speed[e1];
        const float cvm = pg_valid[e1] ? 1.0f : 0.0f;
        penAcc += fmaxf(sQspd[qq] - csl, 0.0f) * cvm * kDT;
      }
    }
    const float oPen = __shfl_xor(penAcc, 16, 32);   // combine lane0 + lane16 deterministically
    penAcc += oPen;
    if (lane == 0) sWavePen[wv] = penAcc;
  }
  __syncthreads();
  if (tid == 0) {
    const float total  = sWavePen[0] + sWavePen[1];
    const float reward = 1.0f - total / (kTHR * (float)T * kDT);
    out[b] = fmaxf(reward, 0.0f);
  }
}

extern "C" void kernel_launch(void* const* d_in, const int* in_sizes, int n_in,
                              void* d_out, int out_size, void* d_ws, size_t ws_size,
                              hipStream_t stream) {
  (void)n_in; (void)out_size; (void)d_ws; (void)ws_size;
  const int*   agent_ptr   = (const int*)  d_in[0];
  const float* aip         = (const float*)d_in[1];
  const float* aih         = (const float*)d_in[2];
  const int*   agent_batch = (const int*)  d_in[3];
  const float* pl_pos      = (const float*)d_in[4];
  const float* pl_head     = (const float*)d_in[5];
  const int*   pl_batch    = (const int*)  d_in[6];
  const int*   pg_type     = (const int*)  d_in[7];
  const float* pg_speed    = (const float*)d_in[8];
  const int*   pg_valid    = (const int*)  d_in[9];
  const int*   edge        = (const int*)  d_in[10];
  float*       out         = (float*)      d_out;

  const int B    = in_sizes[0] - 1;         // 64
  const int Na   = in_sizes[3];             // 2048
  const int Ttot = in_sizes[2] / Na;        // 36
  const int Npl  = in_sizes[5];             // 30000
  const int T    = Ttot - HIST;             // 32
  const int* edge1 = edge + Npl;            // second row of edge_index

  speed_limit_reward_kernel<<<dim3(B), dim3(64), 0, stream>>>(
      agent_ptr, aip, aih, agent_batch, pl_pos, pl_head, pl_batch,
      pg_type, pg_speed, pg_valid, edge1, out, Npl, Ttot, T);
}